// COINSEG_Contrastive_Loss_33414845562971
// MI455X (gfx1250) — compile-verified
//
#include <hip/hip_runtime.h>
#include <math.h>

// ---- problem constants (fixed by the reference) ----
#define BB    8
#define CC    256
#define HH    128
#define WW    128
#define HWSZ  (HH*WW)          // 16384
#define NPIX  (BB*HWSZ)        // 131072
#define FULL  512
#define KNUM  21
#define OLDC  16
#define THRESH 0.7f
#define TEMP   0.07f

#define NBLK  128              // blocks for segment-sum pass
#define SEGF  (2*KNUM*CC)      // 10752 partial-sum floats per block
#define STRIDE 10784           // padded per-block partial stride (sums + 32)
#define NRED  (SEGF + KNUM)    // 10773 reduced elements

typedef float v2f __attribute__((ext_vector_type(2)));
typedef float v8f __attribute__((ext_vector_type(8)));

// ---------------------------------------------------------------------------
// K1: pseudo-label. nearest-down(labels) and, where label==0, first-argmax of
// thresholded outputs_old sampled at (4h,4w).
// ---------------------------------------------------------------------------
__global__ void k_pseudo(const int* __restrict__ labels,
                         const float* __restrict__ outputs_old,
                         int* __restrict__ pseudo) {
  int n = blockIdx.x * 256 + threadIdx.x;
  if (n >= NPIX) return;
  int b = n / HWSZ, hw = n % HWSZ, h = hw / WW, w = hw % WW;
  long long full_off = (long long)b * FULL * FULL + (long long)(4 * h) * FULL + 4 * w;
  int lab = labels[full_off];
  if (lab == 0) {
    const float* po = outputs_old + (long long)b * OLDC * FULL * FULL
                                  + (long long)(4 * h) * FULL + 4 * w;
    float v0 = po[0];
    float best = (v0 < THRESH) ? 0.0f : v0;
    int idx = 0;
    for (int c = 1; c < OLDC; ++c) {
      float v = po[(long long)c * FULL * FULL];
      v = (v < THRESH) ? 0.0f : v;
      if (v > best) { best = v; idx = c; }   // strict > : first-max semantics
    }
    lab = idx;
  }
  pseudo[n] = lab;
}

// ---------------------------------------------------------------------------
// K2: per-pixel inverse L2 norm for both feature tensors.
// 4 consecutive W-pixels per thread -> float4 (16B) coalesced loads.
// This is the primary bandwidth pass (256 MB).
// ---------------------------------------------------------------------------
__global__ void k_invnorm(const float* __restrict__ fa,   // features
                          const float* __restrict__ fo,   // features_old
                          float* __restrict__ inva,
                          float* __restrict__ invo) {
  int g = blockIdx.x * 256 + threadIdx.x;
  int n4 = g * 4;
  if (n4 >= NPIX) return;
  int b = n4 / HWSZ, hw = n4 % HWSZ;     // groups of 4 never cross a batch row
  const float4* pa = (const float4*)(fa + (long long)b * CC * HWSZ + hw);
  const float4* po = (const float4*)(fo + (long long)b * CC * HWSZ + hw);
  float sa0 = 0.f, sa1 = 0.f, sa2 = 0.f, sa3 = 0.f;
  float so0 = 0.f, so1 = 0.f, so2 = 0.f, so3 = 0.f;
#pragma unroll 4
  for (int c = 0; c < CC; ++c) {
    float4 va = pa[(size_t)c * (HWSZ / 4)];
    float4 vo = po[(size_t)c * (HWSZ / 4)];
    sa0 += va.x * va.x; sa1 += va.y * va.y; sa2 += va.z * va.z; sa3 += va.w * va.w;
    so0 += vo.x * vo.x; so1 += vo.y * vo.y; so2 += vo.z * vo.z; so3 += vo.w * vo.w;
  }
  inva[n4 + 0] = 1.0f / fmaxf(sqrtf(sa0), 1e-12f);
  inva[n4 + 1] = 1.0f / fmaxf(sqrtf(sa1), 1e-12f);
  inva[n4 + 2] = 1.0f / fmaxf(sqrtf(sa2), 1e-12f);
  inva[n4 + 3] = 1.0f / fmaxf(sqrtf(sa3), 1e-12f);
  invo[n4 + 0] = 1.0f / fmaxf(sqrtf(so0), 1e-12f);
  invo[n4 + 1] = 1.0f / fmaxf(sqrtf(so1), 1e-12f);
  invo[n4 + 2] = 1.0f / fmaxf(sqrtf(so2), 1e-12f);
  invo[n4 + 3] = 1.0f / fmaxf(sqrtf(so3), 1e-12f);
}

// ---------------------------------------------------------------------------
// K3: segment-sum of normalized features into per-block LDS class accumulators
// (2 x 21 x 256 f32 = 43 KB, fits easily in the 320 KB WGP LDS), flushed to
// per-block global partials (no global atomics).
// ---------------------------------------------------------------------------
__global__ void k_segsum(const float* __restrict__ fa,
                         const float* __restrict__ fo,
                         const int* __restrict__ pseudo,
                         const float* __restrict__ inva,
                         const float* __restrict__ invo,
                         float* __restrict__ partials) {
  __shared__ float lsum[SEGF];
  __shared__ float lcnt[KNUM];
  int tid = threadIdx.x;
  for (int i = tid; i < SEGF; i += 256) lsum[i] = 0.0f;
  if (tid < KNUM) lcnt[tid] = 0.0f;
  __syncthreads();

  int base = blockIdx.x * (NPIX / NBLK);            // 1024 contiguous pixels
  for (int it = 0; it < (NPIX / NBLK) / 256; ++it) {
    int n = base + it * 256 + tid;                  // coalesced across lanes
    int lab = pseudo[n];
    float ia = inva[n], io = invo[n];
    int b = n / HWSZ, hw = n % HWSZ;
    const float* pa = fa + (long long)b * CC * HWSZ + hw;
    const float* po = fo + (long long)b * CC * HWSZ + hw;
    float* da = &lsum[lab * CC];
    float* dn = &lsum[KNUM * CC + lab * CC];
    atomicAdd(&lcnt[lab], 1.0f);
#pragma unroll 4
    for (int c = 0; c < CC; ++c) {
      atomicAdd(&da[c], pa[(size_t)c * HWSZ] * ia);   // ds_add_f32
      atomicAdd(&dn[c], po[(size_t)c * HWSZ] * io);
    }
  }
  __syncthreads();
  float* out = partials + (size_t)blockIdx.x * STRIDE;
  for (int i = tid; i < SEGF; i += 256) out[i] = lsum[i];
  if (tid < KNUM) out[SEGF + tid] = lcnt[tid];
}

// ---------------------------------------------------------------------------
// K4: reduce per-block partials -> [sum_a | sum_o | counts]
// ---------------------------------------------------------------------------
__global__ void k_reduce(const float* __restrict__ partials,
                         float* __restrict__ reduced) {
  int e = blockIdx.x * 256 + threadIdx.x;
  if (e >= NRED) return;
  float s = 0.0f;
  for (int blk = 0; blk < NBLK; ++blk) s += partials[(size_t)blk * STRIDE + e];
  reduced[e] = s;
}

// ---------------------------------------------------------------------------
// K5: final loss. contrast = [anc ; con_prev] lives in one LDS array s_ctr
// (row layout identical to the reduced buffer). adc = anc @ contrast^T / TEMP
// via V_WMMA_F32_16X16X4_F32: 32x48 padded tile grid, one 16x16 tile per wave,
// 64 K=4 steps. Out-of-range rows/cols clamp their index to row 0 instead of
// masking: WMMA outputs are independent dot products, so the garbage lands
// only in adc rows>=21 / cols>=42, which the loss stage never reads. This
// keeps the K-loop branch-free: 2x ds_load_b64 + 1x v_wmma per step, EXEC
// all-1s throughout as WMMA requires.
// ---------------------------------------------------------------------------
__global__ void k_final(const float* __restrict__ reduced,
                        float* __restrict__ out) {
  __shared__ float s_ctr[2 * KNUM * CC];  // 43 KB: rows 0..20 anc, 21..41 con
  __shared__ float s_adc[32 * 48];        // 6 KB
  __shared__ float s_cnt[KNUM];
  __shared__ float s_row[KNUM];
  __shared__ float s_val[KNUM];
  int tid = threadIdx.x;

  if (tid < KNUM) s_cnt[tid] = reduced[SEGF + tid];
  __syncthreads();

  // reduced layout [sum_a(21x256) | sum_o(21x256)] == s_ctr row layout.
  for (int i = tid; i < 2 * KNUM * CC; i += 256) {
    int k = i / CC;                         // contrast row 0..41
    int kc = (k < KNUM) ? k : (k - KNUM);   // class id
    float cnt = s_cnt[kc];
    float inv = (cnt > 0.0f) ? (1.0f / cnt) : 0.0f;
    s_ctr[i] = reduced[i] * inv;
  }
  __syncthreads();

  // All 8 waves execute the WMMA loop (EXEC all-1s); waves 6,7 redundantly
  // recompute tiles 0,1 and store bitwise-identical values (benign).
  int wv = tid / 32;
  int lane = tid % 32;
  int wq = (wv < 6) ? wv : (wv - 6);
  int mt = wq / 3, nt = wq % 3;
  int mloc = lane % 16;
  int half = lane / 16;                 // K-half select per ISA f32 A/B layout
  int row  = mt * 16 + mloc;            // global A row  (valid < 21)
  int colg = nt * 16 + mloc;            // global B col  (valid < 42)
  const float* pa = &s_ctr[((row  <     KNUM) ? row  : 0) * CC];
  const float* pb = &s_ctr[((colg < 2 * KNUM) ? colg : 0) * CC];

  v8f acc = {0.f, 0.f, 0.f, 0.f, 0.f, 0.f, 0.f, 0.f};
  for (int k0 = 0; k0 < CC; k0 += 4) {
    int ka = k0 + 2 * half;
    v2f a, b;
    a.x = pa[ka]; a.y = pa[ka + 1];     // -> ds_load_b64
    b.x = pb[ka]; b.y = pb[ka + 1];     // -> ds_load_b64
    acc = __builtin_amdgcn_wmma_f32_16x16x4_f32(false, a, false, b, (short)0,
                                                acc, false, false);
  }
  const float invT = 1.0f / TEMP;
#pragma unroll
  for (int v = 0; v < 8; ++v) {
    int gr = mt * 16 + v + 8 * half;    // C/D layout: VGPR v, lane-half -> row
    int gc = nt * 16 + mloc;
    s_adc[gr * 48 + gc] = acc[v] * invT;
  }
  __syncthreads();

  if (tid < KNUM) {
    int k = tid;
    float rl = 0.0f, vld = 0.0f;
    if (s_cnt[k] > 0.0f) {
      vld = 1.0f;
      float mx = -1e30f;
      float neg = 0.0f;                 // NOTE: unshifted logits, per reference
      for (int j = 0; j < 2 * KNUM; ++j) {
        if (s_cnt[j % KNUM] > 0.0f) {
          float v = s_adc[k * 48 + j];
          mx = fmaxf(mx, v);
          if (j != k && j != k + KNUM) neg += expf(v);
        }
      }
      float sh = s_adc[k * 48 + (KNUM + k)] - mx;
      float pc = sh - logf(expf(sh) + neg);
      rl = -pc;
    }
    s_row[k] = rl;
    s_val[k] = vld;
  }
  __syncthreads();
  if (tid == 0) {
    float s = 0.0f, nv = 0.0f;
    for (int k = 0; k < KNUM; ++k) { s += s_row[k]; nv += s_val[k]; }
    out[0] = s / fmaxf(nv, 1.0f);
  }
}

// ---------------------------------------------------------------------------
extern "C" void kernel_launch(void* const* d_in, const int* in_sizes, int n_in,
                              void* d_out, int out_size, void* d_ws, size_t ws_size,
                              hipStream_t stream) {
  const int*   labels       = (const int*)  d_in[0];
  const float* features_old = (const float*)d_in[1];
  const float* features     = (const float*)d_in[2];
  const float* outputs_old  = (const float*)d_in[3];
  // d_in[4] outputs, d_in[5] prototypes: dead inputs in the reference.
  float* out = (float*)d_out;

  char* ws = (char*)d_ws;
  int*   pseudo = (int*)ws;                                   //  512 KB
  float* inva   = (float*)(ws + 524288);                      //  512 KB
  float* invo   = (float*)(ws + 1048576);                     //  512 KB
  float* part   = (float*)(ws + 1572864);                     // 5.27 MB
  float* red    = (float*)(ws + 1572864 + (size_t)NBLK * STRIDE * 4);

  k_pseudo <<<NPIX / 256, 256, 0, stream>>>(labels, outputs_old, pseudo);
  k_invnorm<<<(NPIX / 4) / 256, 256, 0, stream>>>(features, features_old, inva, invo);
  k_segsum <<<NBLK, 256, 0, stream>>>(features, features_old, pseudo, inva, invo, part);
  k_reduce <<<(NRED + 255) / 256, 256, 0, stream>>>(part, red);
  k_final  <<<1, 256, 0, stream>>>(red, out);
}